// MLPGNNModel_1142461301287
// MI455X (gfx1250) — compile-verified
//
#include <hip/hip_runtime.h>

// ---------------------------------------------------------------------------
// MLP-GNN collapsed kernel for MI455X (gfx1250, wave32).
//
// pooled[b] = ((1/N) * sum_n w[n] * x[b,n,:]) @ W_gcn + b_gcn
//   w[n] = dinv[n]*(dinv[n] + sum_{e: src_e=n} dinv[dst_e]),  dinv = rsqrt(deg)
// Bandwidth-bound: ~44 MB traffic -> ~1.9 us at 23.3 TB/s.
// Dense head uses V_WMMA_F32_16X16X4_F32 (exact f32 matrix path).
// ---------------------------------------------------------------------------

typedef __attribute__((ext_vector_type(2))) float v2f;
typedef __attribute__((ext_vector_type(8))) float v8f;

#define BATCH 4
#define DHALF 64
#define HDIM  128
#define AS    132   // LDS row stride (HDIM+4): 4-bank skew -> conflict-free

// ---------------- graph-weight pipeline ----------------

__global__ void k_init(float* __restrict__ deg, float* __restrict__ s,
                       float* __restrict__ v, int n) {
  int i = blockIdx.x * blockDim.x + threadIdx.x;
  if (i < n) { deg[i] = 1.0f; s[i] = 0.0f; }   // deg starts at 1 (self-loop)
  if (i < BATCH * HDIM) v[i] = 0.0f;
}

__global__ void k_deg(const int* __restrict__ dst, float* __restrict__ deg, int e) {
  int i = blockIdx.x * blockDim.x + threadIdx.x;
  if (i < e) atomicAdd(&deg[dst[i]], 1.0f);
}

__global__ void k_dinv(const float* __restrict__ deg, float* __restrict__ dinv, int n) {
  int i = blockIdx.x * blockDim.x + threadIdx.x;
  if (i < n) dinv[i] = rsqrtf(deg[i]);         // deg >= 1 always
}

__global__ void k_srcsum(const int* __restrict__ src, const int* __restrict__ dst,
                         const float* __restrict__ dinv, float* __restrict__ s, int e) {
  int i = blockIdx.x * blockDim.x + threadIdx.x;
  if (i < e) atomicAdd(&s[src[i]], dinv[dst[i]]);
}

__global__ void k_weight(const float* __restrict__ dinv, const float* __restrict__ s,
                         float* __restrict__ w, int n, float invN) {
  int i = blockIdx.x * blockDim.x + threadIdx.x;
  if (i < n) { float d = dinv[i]; w[i] = d * (d + s[i]) * invN; }
}

// ---------------- bandwidth-bound weighted feature reduction ----------------
// v[b, 0:128] += sum_n w[n] * concat(feat_a[b,n,:], feat_b[b,n,:])
// 256 threads/block: tid = (slice s in 0..7) * 32 + (group g in 0..31),
// group g owns dims [4g, 4g+4) via one float4 load (coalesced 256B rows).

__global__ void k_wreduce(const float* __restrict__ fa, const float* __restrict__ fb,
                          const float* __restrict__ w, float* __restrict__ v, int n) {
  const int b    = blockIdx.y;
  const int tid  = threadIdx.x;
  const int g    = tid & 31;
  const int sl   = tid >> 5;
  const int gg   = g & 15;
  const float* fsrc = (g < 16) ? fa : fb;      // branch-free pointer select

  float4 acc = make_float4(0.f, 0.f, 0.f, 0.f);
  const int stride = gridDim.x * 8;
  for (int node = blockIdx.x * 8 + sl; node < n; node += stride) {
    const float wn = w[node];
    const float4 x = ((const float4*)(fsrc + ((size_t)b * n + node) * DHALF))[gg];
    acc.x += wn * x.x; acc.y += wn * x.y; acc.z += wn * x.z; acc.w += wn * x.w;
  }

  __shared__ float4 sm[8][32];
  sm[sl][g] = acc;
  __syncthreads();
  if (sl == 0) {
    float4 t = sm[0][g];
#pragma unroll
    for (int j = 1; j < 8; ++j) {
      float4 u = sm[j][g];
      t.x += u.x; t.y += u.y; t.z += u.z; t.w += u.w;
    }
    float* dp = v + b * HDIM + 4 * g;
    atomicAdd(dp + 0, t.x); atomicAdd(dp + 1, t.y);
    atomicAdd(dp + 2, t.z); atomicAdd(dp + 3, t.w);
  }
}

// ---------------- WMMA f32 16x16x4 dense head ----------------
// One 16x16 output tile per wave: Out[0:16, colBase:colBase+16] =
//   Alds(16x128) @ W(128x128 row-major) + bias (optional relu).
// A frag (16x4): lanes 0-15 hold K=k..k+1 of row (lane&15); lanes 16-31 K=k+2..k+3.
// B frag (4x16): lanes 0-15 hold rows k..k+1 at col (colBase+lane&15); lanes 16-31 rows k+2..k+3.
// C/D: VGPR r, lane L -> row r + 8*(L>=16), col L&15.

__device__ __forceinline__ void gemm_tile(const float* __restrict__ Alds,
                                          const float* __restrict__ W,
                                          const float* __restrict__ bias,
                                          float* __restrict__ Out,
                                          int lane, int colBase, bool relu) {
  const int half = lane >> 4;
  const int l16  = lane & 15;
  const int col  = colBase + l16;
  v8f acc = {};
#pragma unroll
  for (int k = 0; k < HDIM; k += 4) {
    const int ka = k + 2 * half;
    v2f a, bm;
    a.x  = Alds[l16 * AS + ka];
    a.y  = Alds[l16 * AS + ka + 1];
    bm.x = W[(size_t)ka * HDIM + col];
    bm.y = W[(size_t)(ka + 1) * HDIM + col];
    acc = __builtin_amdgcn_wmma_f32_16x16x4_f32(
        /*neg_a=*/false, a, /*neg_b=*/false, bm,
        /*c_mod=*/(short)0, acc, /*reuse_a=*/false, /*reuse_b=*/false);
  }
  const float bv = bias[col];
#pragma unroll
  for (int r = 0; r < 8; ++r) {
    const int row = r + 8 * half;
    float val = acc[r] + bv;
    if (relu) val = fmaxf(val, 0.0f);
    Out[row * AS + col] = val;
  }
}

__global__ void __launch_bounds__(256)
k_head(const float* __restrict__ v,
       const float* __restrict__ Wg, const float* __restrict__ bg,
       const float* __restrict__ W1, const float* __restrict__ b1,
       const float* __restrict__ W2, const float* __restrict__ b2,
       float* __restrict__ out) {
  __shared__ float Vp[16 * AS];
  __shared__ float P [16 * AS];
  __shared__ float Hd[16 * AS];

  const int tid = threadIdx.x;
  // Load v into 16x128 tile (rows >= BATCH zero-padded; they only pollute
  // discarded output rows since D-row depends only on the same A-row).
  for (int idx = tid; idx < 16 * HDIM; idx += 256) {
    const int r = idx / HDIM, c = idx % HDIM;
    Vp[r * AS + c] = (r < BATCH) ? v[r * HDIM + c] : 0.0f;
  }
  __syncthreads();

  const int wv = tid >> 5, lane = tid & 31;
  // pooled = v @ W_gcn + b_gcn           (8 waves x one 16-col tile, K=128)
  gemm_tile(Vp, Wg, bg, P, lane, wv * 16, false);
  __syncthreads();
  // hid = relu(pooled @ W1 + b1)
  gemm_tile(P, W1, b1, Hd, lane, wv * 16, true);
  __syncthreads();

  // logits[b] = hid[b,:] . W2 + b2  (wave b of 4 does a shuffle reduction)
  if (tid < BATCH * 32) {
    const int b = tid >> 5, l = tid & 31;
    float p = 0.0f;
#pragma unroll
    for (int d = l; d < HDIM; d += 32) p += Hd[b * AS + d] * W2[d];
#pragma unroll
    for (int off = 16; off > 0; off >>= 1) p += __shfl_down(p, off, 32);
    if (l == 0) out[b] = p + b2[0];
  }
}

// ---------------- launch ----------------

extern "C" void kernel_launch(void* const* d_in, const int* in_sizes, int n_in,
                              void* d_out, int out_size, void* d_ws, size_t ws_size,
                              hipStream_t stream) {
  const float* fa = (const float*)d_in[0];
  const float* fb = (const float*)d_in[1];
  const int*   ei = (const int*)d_in[2];
  const float* Wg = (const float*)d_in[3];
  const float* bg = (const float*)d_in[4];
  const float* W1 = (const float*)d_in[5];
  const float* b1 = (const float*)d_in[6];
  const float* W2 = (const float*)d_in[7];
  const float* b2 = (const float*)d_in[8];
  float* out = (float*)d_out;

  const int E = in_sizes[2] / 2;
  const int N = in_sizes[0] / (BATCH * DHALF);

  float* ws   = (float*)d_ws;
  float* deg  = ws;
  float* dinv = ws + N;
  float* ssum = ws + 2 * (size_t)N;
  float* w    = ws + 3 * (size_t)N;
  float* v    = ws + 4 * (size_t)N;

  const int* src = ei;
  const int* dst = ei + E;

  const int TB = 256;
  k_init  <<<(N + TB - 1) / TB, TB, 0, stream>>>(deg, ssum, v, N);
  k_deg   <<<(E + TB - 1) / TB, TB, 0, stream>>>(dst, deg, E);
  k_dinv  <<<(N + TB - 1) / TB, TB, 0, stream>>>(deg, dinv, N);
  k_srcsum<<<(E + TB - 1) / TB, TB, 0, stream>>>(src, dst, dinv, ssum, E);
  k_weight<<<(N + TB - 1) / TB, TB, 0, stream>>>(dinv, ssum, w, N, 1.0f / (float)N);

  dim3 rgrid(128, BATCH);
  k_wreduce<<<rgrid, 256, 0, stream>>>(fa, fb, w, v, N);

  k_head<<<1, 256, 0, stream>>>(v, Wg, bg, W1, b1, W2, b2, out);
}